// VariationalEncoder_52458730553994
// MI455X (gfx1250) — compile-verified
//
#include <hip/hip_runtime.h>
#include <hip/hip_bf16.h>

typedef __attribute__((ext_vector_type(2))) float v2f;
typedef __attribute__((ext_vector_type(8))) float v8f;

#define IN_DIM  128
#define HID_DIM 128
#define LAT_DIM 32
#define LAT2    64   // z_mean(32) + z_logstd(32) packed
#define LDS_STRIDE 40  // padded: 2*40 mod 64 == 16 -> half-waves hit disjoint banks

// ---------------------------------------------------------------------------
// Utility: fill a float buffer with a constant
// ---------------------------------------------------------------------------
__global__ void vgae_fill_kernel(float* __restrict__ p, float v, int n) {
    int t = blockIdx.x * blockDim.x + threadIdx.x;
    if (t < n) p[t] = v;
}

// deg[dst[e]] += ew[e]   (deg pre-filled with 2.0 for improved self loops)
__global__ void vgae_deg_edge_kernel(const int* __restrict__ dst,
                                     const float* __restrict__ ew,
                                     float* __restrict__ deg, int E) {
    int t = blockIdx.x * blockDim.x + threadIdx.x;
    if (t < E) atomicAdd(&deg[dst[t]], ew[t]);
}

// deg -> dinv in place: dinv = deg > 0 ? rsqrt(deg) : 0
__global__ void vgae_dinv_kernel(float* __restrict__ deg, int N) {
    int t = blockIdx.x * blockDim.x + threadIdx.x;
    if (t < N) {
        float d = deg[t];
        deg[t] = (d > 0.0f) ? rsqrtf(d) : 0.0f;
    }
}

// ---------------------------------------------------------------------------
// Shared GEMM body: one wave computes a 16x32 output tile via two
// V_WMMA_F32_16X16X4_F32 accumulators sharing the A operand.
// B slab (128 K-rows x 32 cols) pre-staged in LDS by the block.
// A layout (16x4 fp32): lane L: M=L&15, half=L>>4; vgprs = A[M][k0+2h .. +1]
// B layout (4x16 fp32): lane L: N=L&15; vgprs = B[k0+2h][N], B[k0+2h+1][N]
// C/D (16x16 fp32): vgpr g: row = g + 8*half, col = L&15
// ---------------------------------------------------------------------------
__device__ __forceinline__ void vgae_gemm_tile32(
    const float* __restrict__ A, const float* __restrict__ bsm,
    float* __restrict__ C, int N, int ldc, int colOut,
    int rowTile, int lane) {
    int rowBase = rowTile << 4;
    int m    = lane & 15;
    int half = lane >> 4;

    int rowA = rowBase + m;
    if (rowA >= N) rowA = N - 1;                 // clamp (dup read; store guarded)
    const float* arow = A + (size_t)rowA * IN_DIM;

    v8f c0 = {}, c1 = {};
    #pragma unroll 4
    for (int k0 = 0; k0 < IN_DIM; k0 += 4) {
        int ka = k0 + (half << 1);
        v2f a = *(const v2f*)(arow + ka);        // 8B contiguous, aligned
        v2f b0, b1;
        b0.x = bsm[ka * LDS_STRIDE + m];
        b0.y = bsm[(ka + 1) * LDS_STRIDE + m];
        b1.x = bsm[ka * LDS_STRIDE + 16 + m];
        b1.y = bsm[(ka + 1) * LDS_STRIDE + 16 + m];
        c0 = __builtin_amdgcn_wmma_f32_16x16x4_f32(
                 false, a, false, b0, (short)0, c0, false, false);
        c1 = __builtin_amdgcn_wmma_f32_16x16x4_f32(
                 false, a, false, b1, (short)0, c1, false, false);
    }

    int colG0 = colOut + m;
    int colG1 = colOut + 16 + m;
    if (rowBase + 16 <= N) {                     // wave-uniform fast path
        #pragma unroll
        for (int g = 0; g < 8; ++g) {
            int row = rowBase + g + (half << 3);
            C[(size_t)row * ldc + colG0] = c0[g];
            C[(size_t)row * ldc + colG1] = c1[g];
        }
    } else {
        #pragma unroll
        for (int g = 0; g < 8; ++g) {
            int row = rowBase + g + (half << 3);
            if (row < N) {
                C[(size_t)row * ldc + colG0] = c0[g];
                C[(size_t)row * ldc + colG1] = c1[g];
            }
        }
    }
}

// GEMM 1: C[N,128] = A[N,128] @ B[128,128]; block = 8 row-tiles x one 32-col pair
__global__ void vgae_gemm1_kernel(const float* __restrict__ A,
                                  const float* __restrict__ B,
                                  float* __restrict__ C, int N) {
    __shared__ float bsm[IN_DIM * LDS_STRIDE];
    const int numColPairs = HID_DIM / 32;                 // 4
    int cp = blockIdx.x % numColPairs;
    int rowBlock = blockIdx.x / numColPairs;
    int colBase = cp * 32;

    for (int idx = threadIdx.x; idx < IN_DIM * 32; idx += blockDim.x) {
        int k = idx >> 5;
        int c = idx & 31;
        bsm[k * LDS_STRIDE + c] = B[(size_t)k * HID_DIM + colBase + c];
    }
    __syncthreads();

    int rowTile = rowBlock * 8 + (threadIdx.x >> 5);
    int numRowTiles = (N + 15) >> 4;
    if (rowTile >= numRowTiles) return;                   // wave-uniform exit
    vgae_gemm_tile32(A, bsm, C, N, HID_DIM, colBase, rowTile, threadIdx.x & 31);
}

// GEMM 2: C[N,64]: cols 0-31 = A@Wm, 32-63 = A@Ws; Wm/Ws are [128,32]
__global__ void vgae_gemm2_kernel(const float* __restrict__ A,
                                  const float* __restrict__ Bm,
                                  const float* __restrict__ Bs,
                                  float* __restrict__ C, int N) {
    __shared__ float bsm[HID_DIM * LDS_STRIDE];
    const int numColPairs = LAT2 / 32;                    // 2
    int cp = blockIdx.x % numColPairs;
    int rowBlock = blockIdx.x / numColPairs;
    const float* B = (cp == 0) ? Bm : Bs;                 // block-uniform select

    for (int idx = threadIdx.x; idx < HID_DIM * 32; idx += blockDim.x) {
        int k = idx >> 5;
        int c = idx & 31;
        bsm[k * LDS_STRIDE + c] = B[(size_t)k * LAT_DIM + c];
    }
    __syncthreads();

    int rowTile = rowBlock * 8 + (threadIdx.x >> 5);
    int numRowTiles = (N + 15) >> 4;
    if (rowTile >= numRowTiles) return;
    vgae_gemm_tile32(A, bsm, C, N, LAT2, cp * 32, rowTile, threadIdx.x & 31);
}

// ---------------------------------------------------------------------------
// Edge scatter, 128 dims: one wave per edge, 4 dims per lane (float4 gather)
// agg[dst] += hlin[src] * (dinv[src]*ew*dinv[dst])
// ---------------------------------------------------------------------------
__global__ void vgae_scatter128_kernel(const float* __restrict__ hlin,
                                       const int* __restrict__ src,
                                       const int* __restrict__ dst,
                                       const float* __restrict__ ew,
                                       const float* __restrict__ dinv,
                                       float* __restrict__ agg, int E) {
    long long t = (long long)blockIdx.x * blockDim.x + threadIdx.x;
    int e = (int)(t >> 5);
    int lane = (int)(t & 31);
    if (e >= E) return;
    int s = src[e], d = dst[e];
    float norm = dinv[s] * ew[e] * dinv[d];
    const float4 v = *(const float4*)(hlin + (size_t)s * HID_DIM + lane * 4);
    float* o = agg + (size_t)d * HID_DIM + lane * 4;
    atomicAdd(o + 0, v.x * norm);
    atomicAdd(o + 1, v.y * norm);
    atomicAdd(o + 2, v.z * norm);
    atomicAdd(o + 3, v.w * norm);
}

// Layer-1 epilogue: h = relu(agg + hlin * 2*dinv^2 + b1)   (in place into agg)
__global__ void vgae_finish1_kernel(float* __restrict__ agg,
                                    const float* __restrict__ hlin,
                                    const float* __restrict__ dinv,
                                    const float* __restrict__ b1, int N) {
    int t = blockIdx.x * blockDim.x + threadIdx.x;
    if (t >= N * HID_DIM) return;
    int i = t >> 7;
    int dch = t & 127;
    float di = dinv[i];
    float v = agg[t] + hlin[t] * (2.0f * di * di) + b1[dch];
    agg[t] = v > 0.0f ? v : 0.0f;
}

// ---------------------------------------------------------------------------
// Edge scatter, 64 dims into d_out: lanes 0-7 -> z_mean block, 8-15 -> z_logstd
// ---------------------------------------------------------------------------
__global__ void vgae_scatter64_kernel(const float* __restrict__ zlin,
                                      const int* __restrict__ src,
                                      const int* __restrict__ dst,
                                      const float* __restrict__ ew,
                                      const float* __restrict__ dinv,
                                      float* __restrict__ out, int N, int E) {
    long long t = (long long)blockIdx.x * blockDim.x + threadIdx.x;
    int e = (int)(t >> 4);
    int lane = (int)(t & 15);
    if (e >= E) return;
    int s = src[e], d = dst[e];
    float norm = dinv[s] * ew[e] * dinv[d];
    int dim0 = lane * 4;                                  // 0..60, never crosses 32
    const float4 v = *(const float4*)(zlin + (size_t)s * LAT2 + dim0);
    float* o = (dim0 < LAT_DIM)
                 ? out + (size_t)d * LAT_DIM + dim0
                 : out + (size_t)N * LAT_DIM + (size_t)d * LAT_DIM + (dim0 - LAT_DIM);
    atomicAdd(o + 0, v.x * norm);
    atomicAdd(o + 1, v.y * norm);
    atomicAdd(o + 2, v.z * norm);
    atomicAdd(o + 3, v.w * norm);
}

// Layer-2 epilogue: add self-loop contribution + biases for both heads
__global__ void vgae_finish2_kernel(float* __restrict__ out,
                                    const float* __restrict__ zlin,
                                    const float* __restrict__ dinv,
                                    const float* __restrict__ bm,
                                    const float* __restrict__ bs, int N) {
    int t = blockIdx.x * blockDim.x + threadIdx.x;
    if (t >= N * LAT_DIM) return;
    int i = t >> 5;
    int j = t & 31;
    float di = dinv[i];
    float g = 2.0f * di * di;
    out[(size_t)i * LAT_DIM + j] += zlin[(size_t)i * LAT2 + j] * g + bm[j];
    out[(size_t)N * LAT_DIM + (size_t)i * LAT_DIM + j]
        += zlin[(size_t)i * LAT2 + LAT_DIM + j] * g + bs[j];
}

// ---------------------------------------------------------------------------
// Host launcher
// ---------------------------------------------------------------------------
extern "C" void kernel_launch(void* const* d_in, const int* in_sizes, int n_in,
                              void* d_out, int out_size, void* d_ws, size_t ws_size,
                              hipStream_t stream) {
    const float* x   = (const float*)d_in[0];
    const int*   ei  = (const int*)  d_in[1];   // [2,E] flat: src then dst
    const float* ew  = (const float*)d_in[2];
    const float* W1  = (const float*)d_in[3];
    const float* b1  = (const float*)d_in[4];
    const float* Wm  = (const float*)d_in[5];
    const float* bm  = (const float*)d_in[6];
    const float* Ws  = (const float*)d_in[7];
    const float* bs  = (const float*)d_in[8];
    float* out = (float*)d_out;

    const int N = in_sizes[0] / IN_DIM;
    const int E = in_sizes[2];
    const int* src = ei;
    const int* dst = ei + E;

    // Workspace layout (floats): [dinv: N][hlin/zlin: 128N][agg/h: 128N]
    float* ws   = (float*)d_ws;
    size_t offN = ((size_t)N + 3) & ~(size_t)3;
    float* dinv = ws;                          // N (deg first, then rsqrt in place)
    float* hlin = ws + offN;                   // 128N (layer1 linear; reused as zlin)
    float* agg  = hlin + (size_t)N * HID_DIM;  // 128N (layer1 aggregate -> h)
    float* zlin = hlin;

    const int TPB = 256;
    auto cdiv = [](long long a, long long b) { return (int)((a + b - 1) / b); };

    // 1) degree: fill 2.0, accumulate edge weights, rsqrt in place
    vgae_fill_kernel<<<cdiv(N, TPB), TPB, 0, stream>>>(dinv, 2.0f, N);
    vgae_deg_edge_kernel<<<cdiv(E, TPB), TPB, 0, stream>>>(dst, ew, dinv, E);
    vgae_dinv_kernel<<<cdiv(N, TPB), TPB, 0, stream>>>(dinv, N);

    // 2) hlin = x @ W1   (WMMA, LDS-staged B)
    {
        int rowTiles  = (N + 15) >> 4;
        int rowBlocks = cdiv(rowTiles, 8);
        vgae_gemm1_kernel<<<rowBlocks * (HID_DIM / 32), TPB, 0, stream>>>(x, W1, hlin, N);
    }

    // 3) propagate layer 1
    vgae_fill_kernel<<<cdiv((long long)N * HID_DIM, TPB), TPB, 0, stream>>>(agg, 0.0f, N * HID_DIM);
    vgae_scatter128_kernel<<<cdiv((long long)E * 32, TPB), TPB, 0, stream>>>(
        hlin, src, dst, ew, dinv, agg, E);
    vgae_finish1_kernel<<<cdiv((long long)N * HID_DIM, TPB), TPB, 0, stream>>>(
        agg, hlin, dinv, b1, N);

    // 4) zlin = h @ [Wm | Ws]   (WMMA, LDS-staged B; writes into freed hlin buffer)
    {
        int rowTiles  = (N + 15) >> 4;
        int rowBlocks = cdiv(rowTiles, 8);
        vgae_gemm2_kernel<<<rowBlocks * (LAT2 / 32), TPB, 0, stream>>>(agg, Wm, Ws, zlin, N);
    }

    // 5) propagate layer 2 directly into d_out (z_mean block, z_logstd block)
    vgae_fill_kernel<<<cdiv((long long)N * LAT2, TPB), TPB, 0, stream>>>(out, 0.0f, N * LAT2);
    vgae_scatter64_kernel<<<cdiv((long long)E * 16, TPB), TPB, 0, stream>>>(
        zlin, src, dst, ew, dinv, out, N, E);
    vgae_finish2_kernel<<<cdiv((long long)N * LAT_DIM, TPB), TPB, 0, stream>>>(
        out, zlin, dinv, bm, bs, N);
}